// GAT_57458072485999
// MI455X (gfx1250) — compile-verified
//
#include <hip/hip_runtime.h>
#include <math.h>

#define Nn 50000
#define Ee 800000
#define ET (Ee + Nn)          // edges + self loops = 850000
#define Bb 64
#define IN_F 30
#define Cc 64
#define Hh 4
#define HC (Hh*Cc)            // 256
#define NCc 2

typedef __attribute__((ext_vector_type(2))) float v2f;
typedef __attribute__((ext_vector_type(8))) float v8f;

__global__ void k_fill(float* __restrict__ p, float v, int n){
  int i = blockIdx.x*blockDim.x + threadIdx.x;
  if(i < n) p[i] = v;
}

// x[N,64] = node_feat[N,30] @ emb_W[30,64] + emb_b
__global__ void k_embed(const float* __restrict__ nf, const float* __restrict__ W,
                        const float* __restrict__ b, float* __restrict__ x){
  int i = blockIdx.x*blockDim.x + threadIdx.x;
  if(i >= Nn*Cc) return;
  int n = i >> 6, c = i & 63;
  float acc = b[c];
  #pragma unroll
  for(int k=0;k<IN_F;k++) acc = fmaf(nf[n*IN_F+k], W[k*Cc+c], acc);
  x[i] = acc;
}

// h[N,256] = x[N,64] @ W[64,256] via V_WMMA_F32_16X16X4_F32 (one wave per 16x16 tile)
__global__ void __launch_bounds__(256) k_feat(const float* __restrict__ x,
                                              const float* __restrict__ W,
                                              float* __restrict__ h){
  int wid  = (int)((blockIdx.x*blockDim.x + threadIdx.x) >> 5);
  int lane = threadIdx.x & 31;
  int tm = wid >> 4, tn = wid & 15;         // 3125 x 16 tiles, exact
  int half = lane >> 4, l16 = lane & 15;
  int m0 = tm*16, n0 = tn*16;
  v8f c = {};
  const float* xrow = x + (size_t)(m0 + l16)*Cc;
  const float* wcol = W + n0 + l16;
  #pragma unroll
  for(int k=0;k<Cc;k+=4){
    // A 16x4 f32: lane holds M=lane%16, K = vgpr + 2*(lane/16)
    v2f a = *(const v2f*)(xrow + k + 2*half);
    // B 4x16 f32: VGPR j holds row K = j + 2*(lane/16), col = lane%16
    v2f b;
    b.x = wcol[(size_t)(k + 2*half    )*HC];
    b.y = wcol[(size_t)(k + 2*half + 1)*HC];
    c = __builtin_amdgcn_wmma_f32_16x16x4_f32(false, a, false, b, (short)0, c, false, false);
  }
  #pragma unroll
  for(int r=0;r<8;r++)  // C/D: VGPR r -> M = r + 8*(lane/16), N = lane%16
    h[(size_t)(m0 + r + 8*half)*HC + n0 + l16] = c[r];
}

__global__ void k_alpha(const float* __restrict__ h, const float* __restrict__ asrc,
                        const float* __restrict__ adst, float* __restrict__ als,
                        float* __restrict__ ald){
  int i = blockIdx.x*blockDim.x + threadIdx.x;
  if(i >= Nn*Hh) return;
  int n = i >> 2, hh = i & 3;
  const float* hp = h + (size_t)n*HC + hh*Cc;
  float s1 = 0.f, s2 = 0.f;
  #pragma unroll 8
  for(int c=0;c<Cc;c++){
    float v = hp[c];
    s1 = fmaf(v, asrc[hh*Cc+c], s1);
    s2 = fmaf(v, adst[hh*Cc+c], s2);
  }
  als[i] = s1; ald[i] = s2;
}

__device__ __forceinline__ void edge_sd(const int* __restrict__ ei, int e, int& src, int& dst){
  if(e < Ee){ src = ei[e]; dst = ei[Ee + e]; }
  else { src = e - Ee; dst = e - Ee; }       // self loops appended
}

__device__ __forceinline__ float lrelu02(float v){ return v > 0.f ? v : 0.2f*v; }

// float atomic max via signed-max / unsigned-min ordering trick
__device__ __forceinline__ void atomicMaxF(float* addr, float val){
  if(val >= 0.f) atomicMax((int*)addr, __float_as_int(val));
  else           atomicMin((unsigned int*)addr, __float_as_uint(val));
}

__global__ void k_edge_max(const int* __restrict__ ei, const float* __restrict__ als,
                           const float* __restrict__ ald, float* __restrict__ m){
  int i = blockIdx.x*blockDim.x + threadIdx.x;
  if(i >= ET*Hh) return;
  int e = i >> 2, hh = i & 3;
  int src, dst; edge_sd(ei, e, src, dst);
  float v = lrelu02(als[src*Hh+hh] + ald[dst*Hh+hh]);
  atomicMaxF(&m[dst*Hh+hh], v);
}

__global__ void k_edge_exp(const int* __restrict__ ei, const float* __restrict__ als,
                           const float* __restrict__ ald, const float* __restrict__ m,
                           float* __restrict__ esc, float* __restrict__ ssum){
  int i = blockIdx.x*blockDim.x + threadIdx.x;
  if(i >= ET*Hh) return;
  int e = i >> 2, hh = i & 3;
  int src, dst; edge_sd(ei, e, src, dst);
  float v = lrelu02(als[src*Hh+hh] + ald[dst*Hh+hh]);
  float ex = expf(v - m[dst*Hh+hh]);
  esc[i] = ex;
  atomicAdd(&ssum[dst*Hh+hh], ex);
}

// one wave per edge: scatter-add a = softmax weight, agg[dst] += h[src]*a
__global__ void __launch_bounds__(256) k_edge_agg(const int* __restrict__ ei,
                                                  const float* __restrict__ hfeat,
                                                  const float* __restrict__ esc,
                                                  const float* __restrict__ ssum,
                                                  float* __restrict__ agg){
  int wid  = (int)((blockIdx.x*blockDim.x + threadIdx.x) >> 5);
  int lane = threadIdx.x & 31;
  int src, dst; edge_sd(ei, wid, src, dst);
  const float* hs = hfeat + (size_t)src*HC;
  float* ad = agg + (size_t)dst*HC;
  #pragma unroll
  for(int it=0; it<8; it++){
    int idx = it*32 + lane;
    int hh = idx >> 6;
    float a = esc[wid*Hh+hh] / (ssum[dst*Hh+hh] + 1e-16f);
    atomicAdd(&ad[idx], hs[idx]*a);
  }
}

__global__ void k_headmean(const float* __restrict__ agg, const float* __restrict__ b,
                           float* __restrict__ x){
  int i = blockIdx.x*blockDim.x + threadIdx.x;
  if(i >= Nn*Cc) return;
  int n = i >> 6, c = i & 63;
  const float* ap = agg + (size_t)n*HC + c;
  x[i] = 0.25f*(ap[0] + ap[64] + ap[128] + ap[192]) + b[c];
}

__global__ void k_pool(const float* __restrict__ x, const int* __restrict__ batch,
                       float* __restrict__ sums, float* __restrict__ cnt){
  int i = blockIdx.x*blockDim.x + threadIdx.x;
  if(i >= Nn*Cc) return;
  int n = i >> 6, c = i & 63;
  int g = batch[n];
  atomicAdd(&sums[g*Cc + c], x[i]);
  if(c == 0) atomicAdd(&cnt[g], 1.0f);
}

// per-graph MLP head + softmax; one thread per graph
__global__ void k_head(const float* __restrict__ sums, const float* __restrict__ cnt,
                       const float* __restrict__ fcW, const float* __restrict__ fcb,
                       const float* __restrict__ oW, const float* __restrict__ ob,
                       float* __restrict__ out){
  int b = threadIdx.x;
  if(b >= Bb) return;
  float cn = cnt[b]; cn = cn < 1.f ? 1.f : cn;
  float g[Cc];
  #pragma unroll 8
  for(int j=0;j<Cc;j++) g[j] = sums[b*Cc+j] / cn;
  float lg0 = ob[0], lg1 = ob[1];
  for(int p=0;p<64;p++){
    float acc = fcb[p];
    #pragma unroll 8
    for(int j=0;j<Cc;j++) acc = fmaf(g[j], fcW[j*64+p], acc);
    acc = acc > 0.f ? acc : 0.01f*acc;       // F.leaky_relu default slope
    lg0 = fmaf(acc, oW[p*NCc+0], lg0);
    lg1 = fmaf(acc, oW[p*NCc+1], lg1);
  }
  float mx = fmaxf(lg0, lg1);
  float e0 = expf(lg0-mx), e1 = expf(lg1-mx);
  float inv = 1.f/(e0+e1);
  out[b*NCc+0] = e0*inv; out[b*NCc+1] = e1*inv;      // proba
  out[Bb*NCc + b*NCc+0] = lg0;                        // logits
  out[Bb*NCc + b*NCc+1] = lg1;
}

static inline int cdiv(long a, long b){ return (int)((a + b - 1)/b); }

extern "C" void kernel_launch(void* const* d_in, const int* in_sizes, int n_in,
                              void* d_out, int out_size, void* d_ws, size_t ws_size,
                              hipStream_t stream){
  (void)in_sizes; (void)n_in; (void)out_size; (void)ws_size;
  const float* node_feat = (const float*)d_in[0];
  const int*   ei        = (const int*)d_in[1];   // [2,E] row-major
  const int*   batch     = (const int*)d_in[2];
  const float* emb_W = (const float*)d_in[4];
  const float* emb_b = (const float*)d_in[5];
  const float* cW[2]    = {(const float*)d_in[6],  (const float*)d_in[10]};
  const float* casrc[2] = {(const float*)d_in[7],  (const float*)d_in[11]};
  const float* cadst[2] = {(const float*)d_in[8],  (const float*)d_in[12]};
  const float* cb[2]    = {(const float*)d_in[9],  (const float*)d_in[13]};
  const float* fcW = (const float*)d_in[14];
  const float* fcb = (const float*)d_in[15];
  const float* oW  = (const float*)d_in[16];
  const float* ob  = (const float*)d_in[17];
  float* out = (float*)d_out;

  float* ws = (float*)d_ws;
  size_t off = 0;
  float* x     = ws + off; off += (size_t)Nn*Cc;   // 3.2M
  float* hfeat = ws + off; off += (size_t)Nn*HC;   // 12.8M
  float* agg   = ws + off; off += (size_t)Nn*HC;   // 12.8M
  float* als   = ws + off; off += (size_t)Nn*Hh;
  float* ald   = ws + off; off += (size_t)Nn*Hh;
  float* mmax  = ws + off; off += (size_t)Nn*Hh;
  float* ssum  = ws + off; off += (size_t)Nn*Hh;
  float* esc   = ws + off; off += (size_t)ET*Hh;   // 3.4M
  float* sums  = ws + off; off += (size_t)Bb*Cc;
  float* cnt   = ws + off; off += (size_t)Bb;

  const int T = 256;

  k_embed<<<cdiv((long)Nn*Cc, T), T, 0, stream>>>(node_feat, emb_W, emb_b, x);

  for(int L=0; L<2; L++){
    k_fill<<<cdiv((long)Nn*HC, T), T, 0, stream>>>(agg, 0.f, Nn*HC);
    k_fill<<<cdiv((long)Nn*Hh, T), T, 0, stream>>>(ssum, 0.f, Nn*Hh);
    k_fill<<<cdiv((long)Nn*Hh, T), T, 0, stream>>>(mmax, -INFINITY, Nn*Hh);
    // 50000 waves (3125 M-tiles x 16 N-tiles), 8 waves/block -> 6250 blocks
    k_feat<<<(Nn/16)*16*32/T, T, 0, stream>>>(x, cW[L], hfeat);
    k_alpha<<<cdiv((long)Nn*Hh, T), T, 0, stream>>>(hfeat, casrc[L], cadst[L], als, ald);
    k_edge_max<<<cdiv((long)ET*Hh, T), T, 0, stream>>>(ei, als, ald, mmax);
    k_edge_exp<<<cdiv((long)ET*Hh, T), T, 0, stream>>>(ei, als, ald, mmax, esc, ssum);
    k_edge_agg<<<ET*32/T, T, 0, stream>>>(ei, hfeat, esc, ssum, agg);  // 106250 blocks, exact
    k_headmean<<<cdiv((long)Nn*Cc, T), T, 0, stream>>>(agg, cb[L], x);
  }

  k_fill<<<cdiv((long)Bb*Cc, T), T, 0, stream>>>(sums, 0.f, Bb*Cc);
  k_fill<<<1, T, 0, stream>>>(cnt, 0.f, Bb);
  k_pool<<<cdiv((long)Nn*Cc, T), T, 0, stream>>>(x, batch, sums, cnt);
  k_head<<<1, Bb, 0, stream>>>(sums, cnt, fcW, fcb, oW, ob, out);
}